// STClustering_32169305047081
// MI455X (gfx1250) — compile-verified
//
#include <hip/hip_runtime.h>
#include <hip/hip_bf16.h>

typedef float v2f __attribute__((ext_vector_type(2)));
typedef float v8f __attribute__((ext_vector_type(8)));

#define NROWS 4096
#define DLAT  32
#define KCL   10
#define BN_EPS 1e-5f

// ---------------------------------------------------------------------------
// Generic C[m][n] = sum_k A[m][k] * W[n][k] + bias[n]  (i.e. A @ W^T + b)
// One wave per 16x16 output tile, V_WMMA_F32_16X16X4_F32, K stepped by 4.
// A-frag: lanes hold row M = lane&15; VGPR k-offset = 2*(lane>>4) + {0,1}
// B-frag: lanes hold col N = lane&15; same K mapping (B = W^T)
// C/D   : VGPR v, row = v + 8*(lane>>4), col = lane&15
// ---------------------------------------------------------------------------
__global__ void gemm_xwt_wmma(const float* __restrict__ A, const float* __restrict__ W,
                              const float* __restrict__ bias, float* __restrict__ C,
                              int M, int N, int K) {
  const int lane  = threadIdx.x & 31;
  const int wave  = threadIdx.x >> 5;
  const int tilesN = N >> 4;
  const int tile  = blockIdx.x * 8 + wave;  // launch sized so tile always valid
  const int tm = (tile / tilesN) << 4;
  const int tn = (tile % tilesN) << 4;
  const int half = lane >> 4;
  const int idx  = lane & 15;

  const float* __restrict__ arow = A + (size_t)(tm + idx) * K;
  const float* __restrict__ wrow = W + (size_t)(tn + idx) * K;

  v8f acc = {0.f, 0.f, 0.f, 0.f, 0.f, 0.f, 0.f, 0.f};
  for (int k0 = 0; k0 < K; k0 += 4) {
    const int ka = k0 + (half << 1);
    v2f a, b;
    a.x = arow[ka];  a.y = arow[ka + 1];
    b.x = wrow[ka];  b.y = wrow[ka + 1];
    acc = __builtin_amdgcn_wmma_f32_16x16x4_f32(false, a, false, b,
                                                (short)0, acc, false, false);
  }

  const float bval = bias[tn + idx];
  const int row0 = tm + (half << 3);
#pragma unroll
  for (int v = 0; v < 8; ++v)
    C[(size_t)(row0 + v) * N + tn + idx] = acc[v] + bval;
}

// ---------------------------------------------------------------------------
// BatchNorm training-mode statistics: one block per feature column.
// ---------------------------------------------------------------------------
__global__ void bn_stats(const float* __restrict__ H, int M, int D,
                         float* __restrict__ mean, float* __restrict__ var) {
  __shared__ float s1[256];
  __shared__ float s2[256];
  const int c = blockIdx.x;
  float a = 0.f, b = 0.f;
  for (int r = threadIdx.x; r < M; r += 256) {
    float h = H[(size_t)r * D + c];
    a += h; b += h * h;
  }
  s1[threadIdx.x] = a; s2[threadIdx.x] = b;
  __syncthreads();
  for (int t = 128; t > 0; t >>= 1) {
    if ((int)threadIdx.x < t) {
      s1[threadIdx.x] += s1[threadIdx.x + t];
      s2[threadIdx.x] += s2[threadIdx.x + t];
    }
    __syncthreads();
  }
  if (threadIdx.x == 0) {
    float mu = s1[0] / (float)M;
    mean[c] = mu;
    var[c]  = s2[0] / (float)M - mu * mu;  // biased variance
  }
}

__global__ void bn_relu(float* __restrict__ H, const float* __restrict__ g,
                        const float* __restrict__ beta, const float* __restrict__ mean,
                        const float* __restrict__ var, int total, int D) {
  const int i = blockIdx.x * 256 + threadIdx.x;
  if (i >= total) return;
  const int c = i & (D - 1);  // D is a power of two (256, 64)
  float h = (H[i] - mean[c]) * rsqrtf(var[c] + BN_EPS) * g[c] + beta[c];
  H[i] = fmaxf(h, 0.f);
}

// ---------------------------------------------------------------------------
// Per-row (one wave32 per row, one lane per latent dim):
//   sum_y, F.normalize, Student-t soft assignment to KCL centers.
// ---------------------------------------------------------------------------
__global__ void latent_post(const float* __restrict__ L, const float* __restrict__ centers,
                            float* __restrict__ sumy, float* __restrict__ dist1,
                            float* __restrict__ dist2, float* __restrict__ normed) {
  const int lane = threadIdx.x & 31;
  const int row  = blockIdx.x * 8 + (threadIdx.x >> 5);
  const float v = L[(size_t)row * DLAT + lane];

  float s = v * v;
#pragma unroll
  for (int m = 16; m > 0; m >>= 1) s += __shfl_xor(s, m, 32);
  if (lane == 0) sumy[row] = s;

  normed[(size_t)row * DLAT + lane] = v / fmaxf(sqrtf(s), 1e-12f);

  float x1[KCL];
  float qc[KCL];
  float qsum = 0.f;
#pragma unroll
  for (int c = 0; c < KCL; ++c) {
    float d = v - centers[c * DLAT + lane];
    d = d * d;
#pragma unroll
    for (int m = 16; m > 0; m >>= 1) d += __shfl_xor(d, m, 32);
    x1[c] = 1.f + d;          // latent_dist2
    qc[c] = 1.f / x1[c];
    qsum += qc[c];
  }
  if (lane == 0) {
#pragma unroll
    for (int c = 0; c < KCL; ++c) {
      dist1[row * KCL + c] = qc[c] / qsum;
      dist2[row * KCL + c] = x1[c];
    }
  }
}

// ---------------------------------------------------------------------------
// num[i][j] = 1 / (1 + sum_y[i] + sum_y[j] - 2 * <latent_i, latent_j>)
// One wave per 16x16 tile of the 4096x4096 affinity, WMMA dot over K=32.
// ---------------------------------------------------------------------------
__global__ void gram_num(const float* __restrict__ L, const float* __restrict__ sumy,
                         float* __restrict__ num) {
  const int lane = threadIdx.x & 31;
  const int wave = threadIdx.x >> 5;
  const int tilesN = NROWS >> 4;                 // 256
  const int tile = blockIdx.x * 8 + wave;
  const int tm = (tile / tilesN) << 4;
  const int tn = (tile % tilesN) << 4;
  const int half = lane >> 4;
  const int idx  = lane & 15;

  const float* __restrict__ ai = L + (size_t)(tm + idx) * DLAT;
  const float* __restrict__ bj = L + (size_t)(tn + idx) * DLAT;

  v8f acc = {0.f, 0.f, 0.f, 0.f, 0.f, 0.f, 0.f, 0.f};
#pragma unroll
  for (int k0 = 0; k0 < DLAT; k0 += 4) {
    const int ka = k0 + (half << 1);
    v2f a, b;
    a.x = ai[ka];  a.y = ai[ka + 1];
    b.x = bj[ka];  b.y = bj[ka + 1];
    acc = __builtin_amdgcn_wmma_f32_16x16x4_f32(false, a, false, b,
                                                (short)0, acc, false, false);
  }

  const float syj = sumy[tn + idx];
  const int row0 = tm + (half << 3);
#pragma unroll
  for (int v = 0; v < 8; ++v) {
    const int i = row0 + v;
    const float d = sumy[i] + syj - 2.f * acc[v];
    num[(size_t)i * NROWS + tn + idx] = 1.f / (1.f + d);
  }
}

// ---------------------------------------------------------------------------
// Row sums of num (num is L2-resident: 64MB < 192MB L2); exact computed
// diagonal captured; Sp = rowsum(zerodiag) = rowsum - diag.
// ---------------------------------------------------------------------------
__global__ void num_rowsum(const float* __restrict__ num, float* __restrict__ Sp,
                           float* __restrict__ diag) {
  __shared__ float red[256];
  const int i = blockIdx.x;
  const float* __restrict__ row = num + (size_t)i * NROWS;
  float s = 0.f;
  for (int j = threadIdx.x; j < NROWS; j += 256) s += row[j];
  red[threadIdx.x] = s;
  __syncthreads();
  for (int t = 128; t > 0; t >>= 1) {
    if ((int)threadIdx.x < t) red[threadIdx.x] += red[threadIdx.x + t];
    __syncthreads();
  }
  if (threadIdx.x == 0) {
    float d = row[i];
    diag[i] = d;
    Sp[i]   = red[0] - d;
  }
}

// ---------------------------------------------------------------------------
// Single fused pass emitting latent_p and latent_q:
//   p = zerodiag/Sp staged in LDS (16KB row buffer), reduce sum(p) & sum(p^2),
//   q = (p^2/Sp1) / (Sp2/Sp1); diagonals patched with diag(num).
// ---------------------------------------------------------------------------
__global__ void pq_pass(const float* __restrict__ num, const float* __restrict__ Sp,
                        const float* __restrict__ diag, float* __restrict__ P,
                        float* __restrict__ Q) {
  __shared__ float prow[NROWS];
  __shared__ float r1[256];
  __shared__ float r2[256];
  const int i = blockIdx.x;
  const float* __restrict__ row = num + (size_t)i * NROWS;
  const float invSp = 1.f / Sp[i];

  float s1 = 0.f, s2 = 0.f;
  for (int j = threadIdx.x; j < NROWS; j += 256) {
    float p = (j == i) ? 0.f : row[j] * invSp;
    prow[j] = p;
    s1 += p; s2 += p * p;
  }
  r1[threadIdx.x] = s1; r2[threadIdx.x] = s2;
  __syncthreads();
  for (int t = 128; t > 0; t >>= 1) {
    if ((int)threadIdx.x < t) {
      r1[threadIdx.x] += r1[threadIdx.x + t];
      r2[threadIdx.x] += r2[threadIdx.x + t];
    }
    __syncthreads();
  }
  const float Sp1 = r1[0];                 // sum(p) (~1.0, reference recomputes it)
  const float Sq  = r2[0] / Sp1;           // rowsum of p^2/Sp1
  const float invSp1 = 1.f / Sp1;
  const float invSq  = 1.f / Sq;
  const float d = diag[i];

  for (int j = threadIdx.x; j < NROWS; j += 256) {
    float p = prow[j];
    float q = (p * p * invSp1) * invSq;
    P[(size_t)i * NROWS + j] = (j == i) ? d : p;   // p + diag
    Q[(size_t)i * NROWS + j] = (j == i) ? d : q;   // q + diag (q_ii == 0)
  }
}

// ---------------------------------------------------------------------------
extern "C" void kernel_launch(void* const* d_in, const int* in_sizes, int n_in,
                              void* d_out, int out_size, void* d_ws, size_t ws_size,
                              hipStream_t stream) {
  (void)in_sizes; (void)n_in; (void)out_size; (void)ws_size;

  const float* x      = (const float*)d_in[0];
  const float* W0     = (const float*)d_in[1];
  const float* b0     = (const float*)d_in[2];
  const float* g0     = (const float*)d_in[3];
  const float* beta0  = (const float*)d_in[4];
  const float* W1     = (const float*)d_in[5];
  const float* b1     = (const float*)d_in[6];
  const float* g1     = (const float*)d_in[7];
  const float* beta1  = (const float*)d_in[8];
  const float* W2     = (const float*)d_in[9];
  const float* b2     = (const float*)d_in[10];
  const float* cent   = (const float*)d_in[11];

  // workspace layout (floats)
  float* ws    = (float*)d_ws;
  float* h0    = ws;                       // 4096*256
  float* h1    = h0 + NROWS * 256;         // 4096*64
  float* mean0 = h1 + NROWS * 64;          // 256
  float* var0  = mean0 + 256;              // 256
  float* mean1 = var0 + 256;               // 64
  float* var1  = mean1 + 64;               // 64
  float* sumy  = var1 + 64;                // 4096
  float* Sp    = sumy + NROWS;             // 4096
  float* dg    = Sp + NROWS;               // 4096

  // output layout (floats), reference tuple order
  float* out     = (float*)d_out;
  float* o_lat   = out;                          // 4096*32
  float* o_d1    = o_lat + NROWS * DLAT;         // 4096*10
  float* o_d2    = o_d1 + NROWS * KCL;           // 4096*10
  float* o_q     = o_d2 + NROWS * KCL;           // 4096*4096
  float* o_p     = o_q + (size_t)NROWS * NROWS;  // 4096*4096
  float* o_num   = o_p + (size_t)NROWS * NROWS;  // 4096*4096
  float* o_norm  = o_num + (size_t)NROWS * NROWS;// 4096*32

  // ---- encoder ----
  // layer 0: 4096x2000 @ 2000x256 -> 4096 tiles -> 512 blocks of 8 waves
  gemm_xwt_wmma<<<512, 256, 0, stream>>>(x, W0, b0, h0, NROWS, 256, 2000);
  bn_stats<<<256, 256, 0, stream>>>(h0, NROWS, 256, mean0, var0);
  bn_relu<<<(NROWS * 256) / 256, 256, 0, stream>>>(h0, g0, beta0, mean0, var0,
                                                   NROWS * 256, 256);
  // layer 1: 4096x256 @ 256x64 -> 1024 tiles -> 128 blocks
  gemm_xwt_wmma<<<128, 256, 0, stream>>>(h0, W1, b1, h1, NROWS, 64, 256);
  bn_stats<<<64, 256, 0, stream>>>(h1, NROWS, 64, mean1, var1);
  bn_relu<<<(NROWS * 64) / 256, 256, 0, stream>>>(h1, g1, beta1, mean1, var1,
                                                  NROWS * 64, 64);
  // layer 2 (bare linear): 4096x64 @ 64x32 -> 512 tiles -> 64 blocks
  gemm_xwt_wmma<<<64, 256, 0, stream>>>(h1, W2, b2, o_lat, NROWS, DLAT, 64);

  // ---- per-row: sum_y, normalize, cluster assignment ----
  latent_post<<<NROWS / 8, 256, 0, stream>>>(o_lat, cent, sumy, o_d1, o_d2, o_norm);

  // ---- N x N affinity via WMMA Gram: 65536 tiles -> 8192 blocks ----
  gram_num<<<8192, 256, 0, stream>>>(o_lat, sumy, o_num);

  // ---- row sums + diagonal of num ----
  num_rowsum<<<NROWS, 256, 0, stream>>>(o_num, Sp, dg);

  // ---- fused p/q emission ----
  pq_pass<<<NROWS, 256, 0, stream>>>(o_num, Sp, dg, o_p, o_q);
}